// Block2x2SelectiveBCDExpertsL2SSM_15934328668594
// MI455X (gfx1250) — compile-verified
//
#include <hip/hip_runtime.h>

// ===========================================================================
// Block2x2 Selective BCD Experts L2 SSM — MI455X (gfx1250, wave32, WMMA)
//
// Pipeline (all on `stream`, in order):
//   1) sigma_kernel   : power iteration -> sigma[m] per expert (8 blocks)
//   2) scale_kernel   : scale = max(max sigma, 1); lambda_p = rho(c+is)/scale
//   3) gate_kernel    : pi = softmax(u @ gate_w^T + b)
//   4) moe_gemm(nm=1) : v = sum_m pi_m * (K12_m/scale) @ u      [WMMA f32]
//   5) scan_local     : per-(b,chunk,pair) local complex scans (chunk=256)
//   6) scan_carry     : carries across chunks (lambda^256 via squaring)
//   7) scan_fix       : z[t] += lambda^j * carry
//   8) moe_gemm(nm=2) : y = sum_m pi_m*(K21_m z + K22_m u)/scale [WMMA f32]
//
// Roofline: ~26 GFLOP f32, ~100 MB DRAM traffic -> memory bound (~5us floor
// at 23.3 TB/s). Keep f32 end-to-end: V_WMMA_F32_16X16X4_F32.
// ===========================================================================

#define EPSR 0.001f

typedef float v2f __attribute__((ext_vector_type(2)));
typedef float v4f __attribute__((ext_vector_type(4)));
typedef float v8f __attribute__((ext_vector_type(8)));

#if defined(__HIP_DEVICE_COMPILE__) && \
    __has_builtin(__builtin_amdgcn_global_load_async_to_lds_b128) && \
    __has_builtin(__builtin_amdgcn_s_wait_asynccnt)
#define USE_ASYNC 1
// Builtin prototype (per hipcc diagnostic): first arg is a pointer to a
// 16-byte int vector; pass address-space-qualified vector pointers.
typedef int v4i_g __attribute__((vector_size(16)));
typedef __attribute__((address_space(1))) v4i_g* gvp_t;
typedef __attribute__((address_space(3))) v4i_g* lvp_t;
#else
#define USE_ASYNC 0
#endif

// ---- problem sizes (fixed by the reference) --------------------------------
constexpr int B_  = 4;
constexpr int T_  = 8192;
constexpr int DX  = 128;
constexpr int DU  = 128;
constexpr int M_  = 8;
constexpr int NP  = 64;            // dx/2 complex pairs
constexpr int BT  = B_ * T_;       // 32768 tokens
constexpr int CHUNK = 256;         // scan chunk length
constexpr int NCH   = T_ / CHUNK;  // 32 chunks per batch row

// ---- workspace layout (in floats) ------------------------------------------
constexpr size_t WS_SIGMA = 0;                                  // 8
constexpr size_t WS_INVS  = 8;                                  // 1
constexpr size_t WS_LAMA  = 16;                                 // 64
constexpr size_t WS_LAMB  = 80;                                 // 64
constexpr size_t WS_PI    = 256;                                // BT*M
constexpr size_t WS_V     = WS_PI  + (size_t)BT * M_;           // BT*DX
constexpr size_t WS_FIN   = WS_V   + (size_t)BT * DX;           // B*NCH*NP*2
constexpr size_t WS_CAR   = WS_FIN + (size_t)B_ * NCH * NP * 2; // B*NCH*NP*2

// ===========================================================================
// 1) Power iteration on K^T K per expert (K built on the fly, 256x256).
//    Row pass: thread r owns row r (float4 contiguous reads).
//    Col pass: thread c owns column c; loop over rows r is lane-coalesced.
// ===========================================================================
__global__ __launch_bounds__(256) void sigma_kernel(
    const float* __restrict__ rho_raw, const float* __restrict__ theta,
    const float* __restrict__ K12, const float* __restrict__ K21,
    const float* __restrict__ K22, float* __restrict__ sigmas)
{
  const int m = blockIdx.x;
  const int t = threadIdx.x;            // 0..255
  __shared__ float xv[256], yv[256], red[256];
  __shared__ float rc[NP], rs[NP];

  if (t < NP) {
    float rho = (1.0f / (1.0f + expf(-rho_raw[t]))) * (1.0f - EPSR);
    rc[t] = rho * cosf(theta[t]);
    rs[t] = rho * sinf(theta[t]);
  }
  xv[t] = 1.0f + 0.0625f * sinf(2.39996f * (float)t);   // non-degenerate start
  __syncthreads();

  const float* k12r = K12 + (size_t)m * DX * DU;
  const float* k21r = K21 + (size_t)m * DX * DX;
  const float* k22r = K22 + (size_t)m * DX * DU;

  const int NIT = 64;
  for (int it = 0; it <= NIT; ++it) {
    // ---- y = K x (row pass) ----
    float acc = 0.0f;
    if (t < 128) {
      int p = t >> 1;
      if ((t & 1) == 0) acc = rc[p] * xv[2 * p] - rs[p] * xv[2 * p + 1];
      else              acc = rs[p] * xv[2 * p] + rc[p] * xv[2 * p + 1];
      const v4f* row = (const v4f*)(k12r + (size_t)t * DU);
      #pragma unroll 4
      for (int c = 0; c < DU / 4; ++c) {
        v4f k = row[c];
        acc += k.x * xv[128 + 4 * c]     + k.y * xv[128 + 4 * c + 1]
             + k.z * xv[128 + 4 * c + 2] + k.w * xv[128 + 4 * c + 3];
      }
    } else {
      int rr = t - 128;
      const v4f* r21 = (const v4f*)(k21r + (size_t)rr * DX);
      const v4f* r22 = (const v4f*)(k22r + (size_t)rr * DU);
      #pragma unroll 4
      for (int c = 0; c < DX / 4; ++c) {
        v4f k = r21[c];
        acc += k.x * xv[4 * c]     + k.y * xv[4 * c + 1]
             + k.z * xv[4 * c + 2] + k.w * xv[4 * c + 3];
      }
      #pragma unroll 4
      for (int c = 0; c < DU / 4; ++c) {
        v4f k = r22[c];
        acc += k.x * xv[128 + 4 * c]     + k.y * xv[128 + 4 * c + 1]
             + k.z * xv[128 + 4 * c + 2] + k.w * xv[128 + 4 * c + 3];
      }
    }
    yv[t] = acc;
    __syncthreads();

    if (it == NIT) {                    // sigma = ||K x_final||, x normalized
      red[t] = acc * acc;
      __syncthreads();
      for (int s = 128; s > 0; s >>= 1) {
        if (t < s) red[t] += red[t + s];
        __syncthreads();
      }
      if (t == 0) sigmas[m] = sqrtf(red[0]);
      return;
    }

    // ---- x' = K^T y (col pass; lanes coalesced over columns) ----
    float a2 = 0.0f;
    if (t < 128) {
      int p = t >> 1;
      if ((t & 1) == 0) a2 =  rc[p] * yv[2 * p] + rs[p] * yv[2 * p + 1];
      else              a2 = -rs[p] * yv[2 * p] + rc[p] * yv[2 * p + 1];
      for (int r = 0; r < DX; ++r) a2 += k21r[(size_t)r * DX + t] * yv[128 + r];
    } else {
      int cc = t - 128;
      for (int r = 0; r < DX; ++r) a2 += k12r[(size_t)r * DU + cc] * yv[r];
      for (int r = 0; r < DX; ++r) a2 += k22r[(size_t)r * DU + cc] * yv[128 + r];
    }
    red[t] = a2 * a2;
    __syncthreads();
    for (int s = 128; s > 0; s >>= 1) {
      if (t < s) red[t] += red[t + s];
      __syncthreads();
    }
    float inv = rsqrtf(red[0] + 1e-30f);
    xv[t] = a2 * inv;
    __syncthreads();
  }
}

// ===========================================================================
// 2) scale + per-pair complex lambda = rho*(cos + i sin)/scale
// ===========================================================================
__global__ void scale_kernel(const float* __restrict__ rho_raw,
                             const float* __restrict__ theta,
                             const float* __restrict__ sigmas,
                             float* __restrict__ ws)
{
  int p = threadIdx.x;                  // 0..63
  float sc = 1.0f;
  for (int m = 0; m < M_; ++m) sc = fmaxf(sc, sigmas[m]);
  float inv = 1.0f / sc;
  if (p == 0) ws[WS_INVS] = inv;
  float rho = (1.0f / (1.0f + expf(-rho_raw[p]))) * (1.0f - EPSR);
  ws[WS_LAMA + p] = rho * cosf(theta[p]) * inv;
  ws[WS_LAMB + p] = rho * sinf(theta[p]) * inv;
}

// ===========================================================================
// 3) gating: pi = softmax(u @ gate_w^T + gate_b)   (GATE_TEMP = 1)
// ===========================================================================
__global__ __launch_bounds__(256) void gate_kernel(
    const float* __restrict__ u, const float* __restrict__ gw,
    const float* __restrict__ gb, float* __restrict__ pi)
{
  __shared__ float gwl[M_][DU];
  for (int i = threadIdx.x; i < M_ * DU; i += 256) gwl[i / DU][i % DU] = gw[i];
  __syncthreads();
  int tok = blockIdx.x * 256 + threadIdx.x;
  const v4f* ur = (const v4f*)(u + (size_t)tok * DU);
  float lg[M_];
  #pragma unroll
  for (int m = 0; m < M_; ++m) lg[m] = gb[m];
  for (int c = 0; c < DU / 4; ++c) {
    v4f uu = ur[c];
    #pragma unroll
    for (int m = 0; m < M_; ++m) {
      lg[m] += uu.x * gwl[m][4 * c]     + uu.y * gwl[m][4 * c + 1]
             + uu.z * gwl[m][4 * c + 2] + uu.w * gwl[m][4 * c + 3];
    }
  }
  float mx = lg[0];
  #pragma unroll
  for (int m = 1; m < M_; ++m) mx = fmaxf(mx, lg[m]);
  float s = 0.0f;
  #pragma unroll
  for (int m = 0; m < M_; ++m) { lg[m] = expf(lg[m] - mx); s += lg[m]; }
  float is = 1.0f / s;
  #pragma unroll
  for (int m = 0; m < M_; ++m) pi[(size_t)tok * M_ + m] = lg[m] * is;
}

// ===========================================================================
// 4/8) MoE GEMM with V_WMMA_F32_16X16X4_F32.
//   out[tok,s] = sum_ph sum_m pi[tok,m]/scale * sum_u K_ph[m,s,u] * x_ph[tok,u]
//   Wave = 16 tokens x 128 outputs (8 N-tiles). A frags (pi-scaled x) live in
//   VGPRs in the documented 16x4 layout. B chunks (128 s x 32 u) are staged to
//   LDS, double-buffered with async loads when the toolchain exposes them.
// ===========================================================================
#define LDSPITCH 36   // dwords per s-row: 16B aligned + conflict-free ds_load_b64

__device__ __forceinline__ void stage_chunk(const float* __restrict__ src,
                                            float* dst, int tid)
{
  #pragma unroll
  for (int i = 0; i < 4; ++i) {
    int e   = tid + 256 * i;        // 0..1023 16-byte units
    int row = e >> 3;               // s row 0..127
    int c16 = e & 7;                // which 16B of the 128B row chunk
    const float* g = src + (size_t)row * DU + c16 * 4;
    float*       l = dst + row * LDSPITCH + c16 * 4;
#if USE_ASYNC
    __builtin_amdgcn_global_load_async_to_lds_b128((gvp_t)g, (lvp_t)l, 0, 0);
#else
    *(v4f*)l = *(const v4f*)g;
#endif
  }
}

__global__ __launch_bounds__(256) void moe_gemm_kernel(
    const float* __restrict__ x1, const float* __restrict__ Km1,
    const float* __restrict__ x2, const float* __restrict__ Km2,
    const float* __restrict__ pi, const float* __restrict__ invs_p,
    float* __restrict__ out, int nmats)
{
  __shared__ float bsm[2][128 * LDSPITCH];
  const int tid  = threadIdx.x;
  const int wave = tid >> 5;
  const int lane = tid & 31;
  const int hi   = lane >> 4;       // A/B K-halves; C/D row halves
  const int l16  = lane & 15;
  const int tokBase = blockIdx.x * 128 + wave * 16;
  const int tok     = tokBase + l16;
  const float invs  = invs_p[0];

  float pis[M_];
  #pragma unroll
  for (int m = 0; m < M_; ++m) pis[m] = pi[(size_t)tok * M_ + m] * invs;

  v8f acc[8];
  #pragma unroll
  for (int n = 0; n < 8; ++n)
    #pragma unroll
    for (int r = 0; r < 8; ++r) acc[n][r] = 0.0f;

  v2f xa[32];                       // 16x128 x-tile in A-fragment layout
  const int NCHUNKS = nmats * 32;   // (phase, m, 4 u-chunks)

  // prologue stage of chunk 0
  stage_chunk(Km1, &bsm[0][0], tid);

  int curPh = -1;
  for (int idx = 0; idx < NCHUNKS; ++idx) {
    int ph = idx >> 5;
    if (ph != curPh) {              // (re)load x tile for this phase
      curPh = ph;
      const float* xp = ph ? x2 : x1;
      #pragma unroll
      for (int j = 0; j < 32; ++j)
        xa[j] = *(const v2f*)(xp + (size_t)tok * DU + 4 * j + 2 * hi);
    }
    if (idx + 1 < NCHUNKS) {        // prefetch next chunk into other buffer
      int nidx = idx + 1;
      int nph = nidx >> 5, nm = (nidx >> 2) & 7, nc = nidx & 3;
      const float* K = nph ? Km2 : Km1;
      stage_chunk(K + (size_t)nm * DX * DU + nc * 32, &bsm[nidx & 1][0], tid);
#if USE_ASYNC
      __builtin_amdgcn_s_wait_asynccnt(4);   // current chunk's 4 loads done
#endif
    } else {
#if USE_ASYNC
      __builtin_amdgcn_s_wait_asynccnt(0);
#endif
    }
    __syncthreads();                // LDS chunk visible to all waves

    int mm = (idx >> 2) & 7;
    int cc = idx & 3;
    float pm = pis[mm];
    const float* bb = &bsm[idx & 1][0];
    #pragma unroll
    for (int j = 0; j < 8; ++j) {
      int jj = cc * 8 + j;
      v2f a;
      a.x = pm * xa[jj].x;          // A frag: lanes 0-15 K={0,1}, 16-31 K={2,3}
      a.y = pm * xa[jj].y;
      int uloc = 4 * j + 2 * hi;
      #pragma unroll
      for (int n = 0; n < 8; ++n) {
        const v2f bf = *(const v2f*)(bb + (n * 16 + l16) * LDSPITCH + uloc);
        acc[n] = __builtin_amdgcn_wmma_f32_16x16x4_f32(
            false, a, false, bf, (short)0, acc[n], false, false);
      }
    }
    __syncthreads();                // done reading before buffer reuse
  }

  // C/D layout: reg r -> token row r (lanes 0-15) / r+8 (lanes 16-31)
  #pragma unroll
  for (int n = 0; n < 8; ++n)
    #pragma unroll
    for (int r = 0; r < 8; ++r)
      out[(size_t)(tokBase + r + 8 * hi) * DX + n * 16 + l16] = acc[n][r];
}

// ===========================================================================
// 5) local chunk scans: z[t0+j] = local_j (local_0 = 0), finals = local_L
//    Recurrence (reference semantics): s_0 = 0, s_{t+1} = lam*s_t + v_t,
//    outputs z[t] = s_t (state BEFORE consuming v_t).
// ===========================================================================
__global__ __launch_bounds__(256) void scan_local_kernel(
    const float* __restrict__ v, const float* __restrict__ ws,
    float* __restrict__ zout, float* __restrict__ finals)
{
  int tid = blockIdx.x * 256 + threadIdx.x;  // 0..8191
  int p  = tid & 63;
  int g  = tid >> 6;                         // (b, chunk)
  int b  = g >> 5;
  int ch = g & 31;
  float la = ws[WS_LAMA + p], lb = ws[WS_LAMB + p];
  float zr = 0.0f, zi = 0.0f;
  size_t base = ((size_t)b * T_ + (size_t)ch * CHUNK) * DX + 2 * p;
  for (int j = 0; j < CHUNK; ++j) {
    v2f zo; zo.x = zr; zo.y = zi;
    *(v2f*)(zout + base + (size_t)j * DX) = zo;
    v2f vv = *(const v2f*)(v + base + (size_t)j * DX);
    float nr = la * zr - lb * zi + vv.x;
    float ni = lb * zr + la * zi + vv.y;
    zr = nr; zi = ni;
  }
  size_t fo = (((size_t)b * NCH + ch) * NP + p) * 2;
  finals[fo] = zr; finals[fo + 1] = zi;
}

// ===========================================================================
// 6) carries across chunks: c_0 = 0, c_{i+1} = lam^CHUNK * c_i + finals_i
// ===========================================================================
__global__ void scan_carry_kernel(const float* __restrict__ ws,
                                  const float* __restrict__ finals,
                                  float* __restrict__ carries)
{
  int tid = threadIdx.x;            // 256 = 4 b * 64 pairs
  int p = tid & 63, b = tid >> 6;
  float la = ws[WS_LAMA + p], lb = ws[WS_LAMB + p];
  float pa = la, pb = lb;           // lam^(2^8) via 8 squarings (CHUNK = 256)
  #pragma unroll
  for (int s = 0; s < 8; ++s) {
    float na = pa * pa - pb * pb;
    float nb = 2.0f * pa * pb;
    pa = na; pb = nb;
  }
  float cr = 0.0f, ci = 0.0f;
  for (int ch = 0; ch < NCH; ++ch) {
    size_t o = (((size_t)b * NCH + ch) * NP + p) * 2;
    carries[o] = cr; carries[o + 1] = ci;
    float fr = finals[o], fi = finals[o + 1];
    float nr = pa * cr - pb * ci + fr;
    float ni = pb * cr + pa * ci + fi;
    cr = nr; ci = ni;
  }
}

// ===========================================================================
// 7) fix-up: z[t0+j] += lam^j * carry
// ===========================================================================
__global__ __launch_bounds__(256) void scan_fix_kernel(
    const float* __restrict__ ws, const float* __restrict__ carries,
    float* __restrict__ zout)
{
  int tid = blockIdx.x * 256 + threadIdx.x;
  int p  = tid & 63;
  int g  = tid >> 6;
  int b  = g >> 5;
  int ch = g & 31;
  float la = ws[WS_LAMA + p], lb = ws[WS_LAMB + p];
  size_t fo = (((size_t)b * NCH + ch) * NP + p) * 2;
  float pr = carries[fo], pj = carries[fo + 1];   // lam^0 * c
  size_t base = ((size_t)b * T_ + (size_t)ch * CHUNK) * DX + 2 * p;
  for (int j = 0; j < CHUNK; ++j) {
    v2f z = *(v2f*)(zout + base + (size_t)j * DX);
    z.x += pr; z.y += pj;
    *(v2f*)(zout + base + (size_t)j * DX) = z;
    float nr = la * pr - lb * pj;
    float nj = lb * pr + la * pj;
    pr = nr; pj = nj;
  }
}

// ===========================================================================
// launch
// ===========================================================================
extern "C" void kernel_launch(void* const* d_in, const int* in_sizes, int n_in,
                              void* d_out, int out_size, void* d_ws, size_t ws_size,
                              hipStream_t stream)
{
  const float* u       = (const float*)d_in[0];
  const float* rho_raw = (const float*)d_in[1];
  const float* theta   = (const float*)d_in[2];
  const float* K12     = (const float*)d_in[3];
  const float* K21     = (const float*)d_in[4];
  const float* K22     = (const float*)d_in[5];
  const float* gate_w  = (const float*)d_in[6];
  const float* gate_b  = (const float*)d_in[7];

  float* ws = (float*)d_ws;
  float* y  = (float*)d_out;                 // (B,T,dy)
  float* z  = y + (size_t)BT * DX;           // (B,T,dx)

  sigma_kernel<<<M_, 256, 0, stream>>>(rho_raw, theta, K12, K21, K22,
                                       ws + WS_SIGMA);
  scale_kernel<<<1, 64, 0, stream>>>(rho_raw, theta, ws + WS_SIGMA, ws);
  gate_kernel<<<BT / 256, 256, 0, stream>>>(u, gate_w, gate_b, ws + WS_PI);
  moe_gemm_kernel<<<BT / 128, 256, 0, stream>>>(u, K12, u, K12,
                                                ws + WS_PI, ws + WS_INVS,
                                                ws + WS_V, 1);
  scan_local_kernel<<<(B_ * NCH * NP) / 256, 256, 0, stream>>>(
      ws + WS_V, ws, z, ws + WS_FIN);
  scan_carry_kernel<<<1, 256, 0, stream>>>(ws, ws + WS_FIN, ws + WS_CAR);
  scan_fix_kernel<<<(B_ * NCH * NP) / 256, 256, 0, stream>>>(
      ws, ws + WS_CAR, z);
  moe_gemm_kernel<<<BT / 128, 256, 0, stream>>>(z, K21, u, K22,
                                                ws + WS_PI, ws + WS_INVS,
                                                y, 2);
}